// GaussianGNN_22789096472978
// MI455X (gfx1250) — compile-verified
//
#include <hip/hip_runtime.h>
#include <math.h>

// ---------------------------------------------------------------------------
// GaussianGNN (2-layer single-head GAT + linear readout) for gfx1250 (MI455X)
// Dense GEMMs via V_WMMA_F32_16X16X4_F32 (full f32 precision, wave32 tiles),
// W staged in LDS (320KB/WGP), 4 WMMAs per A-fragment (16x64 strip per wave).
// Edge softmax/aggregation via atomics (features L2-resident, 192MB L2).
// ---------------------------------------------------------------------------

typedef __attribute__((ext_vector_type(2))) float v2f;
typedef __attribute__((ext_vector_type(8))) float v8f;

#define F_IN 128
#define HID  64

// ---------------- ordered-uint encoding for float atomicMax ---------------
__device__ __forceinline__ unsigned enc_f32(float f) {
  unsigned u = __float_as_uint(f);
  return (u & 0x80000000u) ? ~u : (u | 0x80000000u);
}
__device__ __forceinline__ float dec_f32(unsigned e) {
  unsigned u = (e & 0x80000000u) ? (e & 0x7FFFFFFFu) : ~e;
  return __uint_as_float(u);
}
// enc(-inf): u = 0xFF800000 -> ~u = 0x007FFFFF
#define ENC_NEG_INF 0x007FFFFFu

__device__ __forceinline__ float lrelu(float v, float slope) {
  return v > 0.0f ? v : v * slope;
}

// ---------------- wd[k] = sum_j W1_dst[k][j] * a1_dst[j] -------------------
__global__ void make_wd_kernel(const float* __restrict__ Wdst,
                               const float* __restrict__ adst,
                               float* __restrict__ wd) {
  int k = threadIdx.x;            // 128 threads
  float s = 0.0f;
  const float* row = Wdst + (size_t)k * HID;
#pragma unroll 8
  for (int j = 0; j < HID; ++j) s += row[j] * adst[j];
  wd[k] = s;
}

// ---------------- C[M x 64] = A[M x K] @ W[K x 64] via f32 WMMA ------------
// Block: 256 threads = 8 waves; block tile: 128 rows x 64 cols.
// W [K x 64] staged once in LDS; each wave computes a 16x64 strip:
// one A fragment (16x4 f32) feeds 4 independent WMMAs per k-step.
// A-frag lane layout (16x4 f32): lane l holds row M=l&15, K = k0 + (l>>4)*2 (+1).
// B-frag lane layout (4x16 f32): lane l holds col N=l&15, K = k0 + (l>>4)*2 (+1).
// C/D layout (16x16 f32): lane l col N=l&15, VGPR r -> row M = r + (l>>4)*8.
template <int K>
__global__ void gemm_wmma_f32(const float* __restrict__ A,
                              const float* __restrict__ W,
                              float* __restrict__ C, int M) {
  __shared__ float sW[K * HID];

  const int tid = threadIdx.x;
  // cooperative LDS stage of W (K*64 floats, float4 per thread per trip)
  for (int i = tid * 4; i < K * HID; i += 256 * 4)
    *(float4*)&sW[i] = *(const float4*)&W[i];
  __syncthreads();

  const int lane = tid & 31;
  const int wave = tid >> 5;
  const int row0 = blockIdx.x * 128 + wave * 16;
  const int mr = lane & 15;
  const int kh = (lane >> 4) * 2;

  int rowA = row0 + mr;
  int rowAc = rowA < M ? rowA : (M - 1);   // clamp (EXEC must stay all-ones)
  const float* ap = A + (size_t)rowAc * K + kh;

  v8f acc0 = {}, acc1 = {}, acc2 = {}, acc3 = {};
#pragma unroll
  for (int k0 = 0; k0 < K; k0 += 4) {
    if (k0 + 16 < K) __builtin_prefetch(ap + 16, 0, 3);   // WGP-scope prefetch
    v2f a;
    a.x = ap[0];
    a.y = ap[1];
    const float* wp = &sW[(k0 + kh) * HID + mr];
    v2f b0, b1, b2, b3;
    b0.x = wp[0];       b0.y = wp[HID + 0];
    b1.x = wp[16];      b1.y = wp[HID + 16];
    b2.x = wp[32];      b2.y = wp[HID + 32];
    b3.x = wp[48];      b3.y = wp[HID + 48];
    acc0 = __builtin_amdgcn_wmma_f32_16x16x4_f32(false, a, false, b0, (short)0, acc0, false, false);
    acc1 = __builtin_amdgcn_wmma_f32_16x16x4_f32(false, a, false, b1, (short)0, acc1, false, false);
    acc2 = __builtin_amdgcn_wmma_f32_16x16x4_f32(false, a, false, b2, (short)0, acc2, false, false);
    acc3 = __builtin_amdgcn_wmma_f32_16x16x4_f32(false, a, false, b3, (short)0, acc3, false, false);
    ap += 4;
  }

  const int ncol = lane & 15;
  const int mbase = row0 + (lane >> 4) * 8;
#pragma unroll
  for (int r = 0; r < 8; ++r) {
    int mrow = mbase + r;
    if (mrow < M) {
      float* crow = C + (size_t)mrow * HID + ncol;
      crow[0]  = acc0[r];
      crow[16] = acc1[r];
      crow[32] = acc2[r];
      crow[48] = acc3[r];
    }
  }
}

// ---------------- per-node attention logits --------------------------------
// as[i] = dot(S[i, :sLen], sv) ; ad[i] = dot(D[i, :dLen], dv)
__global__ void alphas_kernel(const float* __restrict__ S, const float* __restrict__ sv,
                              int sLen,
                              const float* __restrict__ D, const float* __restrict__ dv,
                              int dLen,
                              float* __restrict__ as_, float* __restrict__ ad_, int N) {
  int i = blockIdx.x * blockDim.x + threadIdx.x;
  if (i >= N) return;
  const float4* srow = (const float4*)(S + (size_t)i * sLen);
  const float4* svv = (const float4*)sv;
  float s = 0.0f;
  for (int j = 0; j < (sLen >> 2); ++j) {
    float4 a = srow[j], b = svv[j];
    s += a.x * b.x + a.y * b.y + a.z * b.z + a.w * b.w;
  }
  const float4* drow = (const float4*)(D + (size_t)i * dLen);
  const float4* dvv = (const float4*)dv;
  float d = 0.0f;
  for (int j = 0; j < (dLen >> 2); ++j) {
    float4 a = drow[j], b = dvv[j];
    d += a.x * b.x + a.y * b.y + a.z * b.z + a.w * b.w;
  }
  as_[i] = s;
  ad_[i] = d;
}

// ---------------- init m / denom / agg -------------------------------------
__global__ void init_kernel(unsigned* __restrict__ m, float* __restrict__ denom,
                            float* __restrict__ agg, int N) {
  int i = blockIdx.x * blockDim.x + threadIdx.x;
  if (i < N) {
    m[i] = ENC_NEG_INF;
    denom[i] = 0.0f;
  }
  if (i < N * HID) agg[i] = 0.0f;
}

// ---------------- pass 1: segment max over dst -----------------------------
__global__ void edge_max_kernel(const long long* __restrict__ ei, long long E, int N,
                                const float* __restrict__ as_, const float* __restrict__ ad_,
                                unsigned* __restrict__ m) {
  long long e = (long long)blockIdx.x * blockDim.x + threadIdx.x;
  long long ET = E + N;
  if (e >= ET) return;
  int src, dst;
  if (e < E) { src = (int)ei[e]; dst = (int)ei[E + e]; }
  else       { src = (int)(e - E); dst = src; }
  float v = lrelu(as_[src] + ad_[dst], 0.2f);
  atomicMax(&m[dst], enc_f32(v));
}

// ---------------- pass 2: exp + segment sum --------------------------------
__global__ void edge_exp_kernel(const long long* __restrict__ ei, long long E, int N,
                                const float* __restrict__ as_, const float* __restrict__ ad_,
                                const unsigned* __restrict__ m,
                                float* __restrict__ denom, float* __restrict__ eexp) {
  long long e = (long long)blockIdx.x * blockDim.x + threadIdx.x;
  long long ET = E + N;
  if (e >= ET) return;
  int src, dst;
  if (e < E) { src = (int)ei[e]; dst = (int)ei[E + e]; }
  else       { src = (int)(e - E); dst = src; }
  float v = lrelu(as_[src] + ad_[dst], 0.2f);
  float ex = __expf(v - dec_f32(m[dst]));
  eexp[e] = ex;
  atomicAdd(&denom[dst], ex);
}

// ---------------- pass 3: weighted scatter-add of xs[src] ------------------
// 16 threads per edge, 4 features (one float4 gather) per thread.
__global__ void edge_agg_kernel(const long long* __restrict__ ei, long long E, int N,
                                const float* __restrict__ eexp,
                                const float* __restrict__ denom,
                                const float* __restrict__ xs,
                                float* __restrict__ agg) {
  long long t = (long long)blockIdx.x * blockDim.x + threadIdx.x;
  long long ET = E + N;
  long long e = t >> 4;
  if (e >= ET) return;
  int c4 = ((int)t & 15) * 4;
  int src, dst;
  if (e < E) { src = (int)ei[e]; dst = (int)ei[E + e]; }
  else       { src = (int)(e - E); dst = src; }
  float attn = eexp[e] / denom[dst];
  float4 v = *(const float4*)(xs + (size_t)src * HID + c4);
  float* o = agg + (size_t)dst * HID + c4;
  atomicAdd(o + 0, attn * v.x);
  atomicAdd(o + 1, attn * v.y);
  atomicAdd(o + 2, attn * v.z);
  atomicAdd(o + 3, attn * v.w);
}

// ---------------- bias (+ optional leaky relu, slope=1 -> identity) --------
__global__ void bias_act_kernel(float* __restrict__ h, const float* __restrict__ b,
                                int N, float slope) {
  int i = blockIdx.x * blockDim.x + threadIdx.x;
  if (i >= N * HID) return;
  float v = h[i] + b[i & (HID - 1)];
  h[i] = lrelu(v, slope);
}

// ---------------- readout: mean / softplus variance ------------------------
__global__ void readout_kernel(const float* __restrict__ h, const float* __restrict__ Wl,
                               const float* __restrict__ bl, float* __restrict__ out,
                               int N) {
  int i = blockIdx.x * blockDim.x + threadIdx.x;
  if (i >= N) return;
  const float* row = h + (size_t)i * HID;
  float o0 = bl[0], o1 = bl[1];
#pragma unroll 8
  for (int j = 0; j < HID; ++j) {
    float v = row[j];
    o0 += v * Wl[2 * j + 0];
    o1 += v * Wl[2 * j + 1];
  }
  out[i] = o0;                                                // mean
  float sp = fmaxf(o1, 0.0f) + log1pf(expf(-fabsf(o1)));      // softplus
  out[(size_t)N + i] = sp + 1e-6f;                            // variance
}

// ---------------------------------------------------------------------------
extern "C" void kernel_launch(void* const* d_in, const int* in_sizes, int n_in,
                              void* d_out, int out_size, void* d_ws, size_t ws_size,
                              hipStream_t stream) {
  const float*     x   = (const float*)d_in[0];
  const long long* ei  = (const long long*)d_in[1];
  const float*     W1s = (const float*)d_in[2];
  const float*     W1d = (const float*)d_in[3];
  const float*     a1s = (const float*)d_in[4];
  const float*     a1d = (const float*)d_in[5];
  const float*     b1  = (const float*)d_in[6];
  const float*     W2  = (const float*)d_in[7];
  const float*     a2s = (const float*)d_in[8];
  const float*     a2d = (const float*)d_in[9];
  const float*     b2  = (const float*)d_in[10];
  const float*     Wl  = (const float*)d_in[11];
  const float*     bl  = (const float*)d_in[12];

  const int       N  = in_sizes[0] / F_IN;
  const long long E  = (long long)in_sizes[1] / 2;
  const long long ET = E + N;

  // ---- workspace carve ----
  char* p = (char*)d_ws;
  float* A     = (float*)p;  p += (size_t)N * HID * sizeof(float);  // xs buffer
  float* B     = (float*)p;  p += (size_t)N * HID * sizeof(float);  // agg / hidden
  float* as_   = (float*)p;  p += (size_t)N * sizeof(float);
  float* ad_   = (float*)p;  p += (size_t)N * sizeof(float);
  unsigned* m  = (unsigned*)p; p += (size_t)N * sizeof(unsigned);
  float* denom = (float*)p;  p += (size_t)N * sizeof(float);
  float* eexp  = (float*)p;  p += (size_t)ET * sizeof(float);
  float* wd    = (float*)p;  p += 128 * sizeof(float);

  const int T = 256;
  const int gN    = (N + T - 1) / T;
  const int gNF   = (N * HID + T - 1) / T;
  const int gE    = (int)((ET + T - 1) / T);
  const int gE16  = (int)((ET * 16 + T - 1) / T);
  const int gGemm = (N + 127) / 128;

  // ================= Layer 1 =================
  make_wd_kernel<<<1, 128, 0, stream>>>(W1d, a1d, wd);
  gemm_wmma_f32<F_IN><<<gGemm, T, 0, stream>>>(x, W1s, A, N);            // xs1
  alphas_kernel<<<gN, T, 0, stream>>>(A, a1s, HID, x, wd, F_IN, as_, ad_, N);
  init_kernel<<<gNF, T, 0, stream>>>(m, denom, B, N);
  edge_max_kernel<<<gE, T, 0, stream>>>(ei, E, N, as_, ad_, m);
  edge_exp_kernel<<<gE, T, 0, stream>>>(ei, E, N, as_, ad_, m, denom, eexp);
  edge_agg_kernel<<<gE16, T, 0, stream>>>(ei, E, N, eexp, denom, A, B);
  bias_act_kernel<<<gNF, T, 0, stream>>>(B, b1, N, 0.01f);               // h1

  // ================= Layer 2 (W shared: alpha_d2 = xs2 . a2_dst) ==========
  gemm_wmma_f32<HID><<<gGemm, T, 0, stream>>>(B, W2, A, N);              // xs2
  alphas_kernel<<<gN, T, 0, stream>>>(A, a2s, HID, A, a2d, HID, as_, ad_, N);
  init_kernel<<<gNF, T, 0, stream>>>(m, denom, B, N);
  edge_max_kernel<<<gE, T, 0, stream>>>(ei, E, N, as_, ad_, m);
  edge_exp_kernel<<<gE, T, 0, stream>>>(ei, E, N, as_, ad_, m, denom, eexp);
  edge_agg_kernel<<<gE16, T, 0, stream>>>(ei, E, N, eexp, denom, A, B);
  bias_act_kernel<<<gNF, T, 0, stream>>>(B, b2, N, 1.0f);                // h2 (no act)

  // ================= Readout =================
  readout_kernel<<<gN, T, 0, stream>>>(B, Wl, bl, (float*)d_out, N);
}